// SimpleGAT_76536317215219
// MI455X (gfx1250) — compile-verified
//
#include <hip/hip_runtime.h>
#include <math.h>

// ---------------------------------------------------------------------------
// SimpleGAT (3x GATConv, heads=1) for MI455X / gfx1250.
//  - Dense per-node linear transforms via V_WMMA_F32_16X16X4_F32 (exact f32),
//    with A-tile + W staged into LDS via global_load_async_to_lds_b128
//    (CDNA5 async copy, ASYNCcnt) and bank-conflict-free padded LDS reads.
//  - Edge softmax via 3 streaming passes with native f32 atomics; segment max
//    uses an order-preserving u32 encoding + integer atomicMax (no CAS loop).
// Assumes N % 32 == 0 (harness: N = 100000).
// ---------------------------------------------------------------------------

typedef __attribute__((ext_vector_type(2))) float v2f;
typedef __attribute__((ext_vector_type(8))) float v8f;

#define DH 64
#define NG 128
#define NC 10
#define TM 32                      // rows per GEMM block

// Order-preserving float <-> u32 total-order map (for integer atomicMax).
__device__ __forceinline__ unsigned enc_f32(float f) {
  unsigned u = __float_as_uint(f);
  return (u & 0x80000000u) ? ~u : (u | 0x80000000u);
}
__device__ __forceinline__ float dec_f32(unsigned u) {
  unsigned v = (u & 0x80000000u) ? (u & 0x7FFFFFFFu) : ~u;
  return __uint_as_float(v);
}

__device__ __forceinline__ void async_copy16(unsigned lds_off, const void* gaddr) {
  asm volatile("global_load_async_to_lds_b128 %0, %1, off"
               :: "v"(lds_off), "v"((unsigned long long)(uintptr_t)gaddr)
               : "memory");
}
__device__ __forceinline__ void wait_async0() {
  asm volatile("s_wait_asynccnt 0x0" ::: "memory");
}

// ---------------------------------------------------------------------------
// Layer-1 linear: h = [pos | x | 0pad] @ W1, K padded 3->4. One wave / 16x16
// tile, single 16x16x4 f32 WMMA. Grid is exact (n % 16 == 0): no guards.
// A layout (32b A 16x4): lanes 0-15 row M, v0=K0 v1=K1; lanes 16-31 v0=K2 v1=K3.
// B layout (32b B 4x16): lanes 0-15 col N hold K0(v0),K1(v1); lanes 16-31 K2,K3.
// ---------------------------------------------------------------------------
__global__ __launch_bounds__(32) void gat_linear1_wmma(
    const float* __restrict__ pos, const float* __restrict__ x,
    const float* __restrict__ W, float* __restrict__ hout)
{
  const int lane = threadIdx.x;
  const int half = lane >> 4;
  const int lid  = lane & 15;
  const int rowBase = blockIdx.x * 16;
  const int colBase = blockIdx.y * 16;
  const int row = rowBase + lid;
  const int col = colBase + lid;

  v2f a, b;
  if (half == 0) { a.x = pos[2 * row];    a.y = pos[2 * row + 1]; }  // K0,K1
  else           { a.x = x[row];          a.y = 0.0f;            }   // K2,K3
  if (half == 0) { b.x = W[0 * DH + col]; b.y = W[1 * DH + col]; }   // K0,K1
  else           { b.x = W[2 * DH + col]; b.y = 0.0f;            }   // K2,K3

  v8f c = {};
  c = __builtin_amdgcn_wmma_f32_16x16x4_f32(false, a, false, b,
                                            (short)0, c, false, false);
  #pragma unroll
  for (int j = 0; j < 8; ++j)
    hout[(size_t)(rowBase + j + 8 * half) * DH + col] = c[j];
}

// ---------------------------------------------------------------------------
// Layers 2/3 linear: h = (h_agg_prev + b_prev) @ W  (bias fused at A read).
// 256 threads = 8 waves per block; block computes a 32x64 tile.
//   - W (64x64 f32, 16KB) and A tile (32x64, 8.3KB padded) staged to LDS via
//     async b128 copies; bias staged via plain LDS stores.
//   - Wave w: row strip (w>>2)*16, col tile (w&3)*16; 16 chained f32 WMMAs.
// ---------------------------------------------------------------------------
__global__ __launch_bounds__(256) void gat_linear_wmma(
    const float* __restrict__ hin, const float* __restrict__ bias_in,
    const float* __restrict__ W, float* __restrict__ hout)
{
  __shared__ float sA[TM][DH + 1];   // +1 pad: conflict-free column reads
  __shared__ float sW[DH][DH + 1];
  __shared__ float sB[DH];

  const int t = threadIdx.x;
  const size_t rowBase = (size_t)blockIdx.x * TM;

  // Stage W: 64 rows x 4 chunks of 16B = 256 chunks, one per thread.
  {
    const int r = t >> 2, cc = (t & 3) * 4;
    async_copy16((unsigned)(uintptr_t)&sW[r][cc], W + (size_t)r * DH + cc);
  }
  // Stage A tile: 32 rows x 4 chunks = 128 chunks (threads 0..127).
  if (t < TM * 4) {
    const int r = t >> 2, cc = (t & 3) * 4;
    async_copy16((unsigned)(uintptr_t)&sA[r][cc],
                 hin + (rowBase + r) * DH + cc);
  }
  // Stage bias (threads 128..191).
  if (t >= 128 && t < 128 + DH) sB[t - 128] = bias_in[t - 128];

  wait_async0();
  __syncthreads();

  const int wave = t >> 5;
  const int lane = t & 31;
  const int half = lane >> 4;
  const int lid  = lane & 15;
  const int strip = (wave >> 2) * 16;   // 0 or 16
  const int colB  = (wave & 3) * 16;    // 0,16,32,48

  v8f c = {};
  #pragma unroll
  for (int k0 = 0; k0 < DH; k0 += 4) {
    const int ka = k0 + 2 * half;       // this lane-half's K pair
    v2f a, b;
    a.x = sA[strip + lid][ka]     + sB[ka];
    a.y = sA[strip + lid][ka + 1] + sB[ka + 1];
    b.x = sW[ka][colB + lid];
    b.y = sW[ka + 1][colB + lid];
    c = __builtin_amdgcn_wmma_f32_16x16x4_f32(false, a, false, b,
                                              (short)0, c, false, false);
  }

  #pragma unroll
  for (int j = 0; j < 8; ++j)
    hout[(rowBase + strip + j + 8 * half) * DH + colB + lid] = c[j];
}

// ---------------------------------------------------------------------------
// Per-node prep: alpha_s = h . a_src, alpha_d = h . a_dst; init segment max
// (encoded -inf), segment sum = 0, zero the aggregation buffer row.
// ---------------------------------------------------------------------------
__global__ void gat_node_prep(
    const float* __restrict__ h,
    const float* __restrict__ a_src, const float* __restrict__ a_dst,
    float* __restrict__ as, float* __restrict__ ad,
    unsigned* __restrict__ menc, float* __restrict__ ssum,
    float* __restrict__ hagg, int n)
{
  int i = blockIdx.x * blockDim.x + threadIdx.x;
  if (i >= n) return;
  const float4* hv = (const float4*)(h + (size_t)i * DH);
  const float4* sv = (const float4*)a_src;
  const float4* dv = (const float4*)a_dst;
  float4* agg = (float4*)(hagg + (size_t)i * DH);
  float s = 0.f, d = 0.f;
  #pragma unroll
  for (int k = 0; k < DH / 4; ++k) {
    float4 hh = hv[k], sa = sv[k], da = dv[k];
    s += hh.x * sa.x + hh.y * sa.y + hh.z * sa.z + hh.w * sa.w;
    d += hh.x * da.x + hh.y * da.y + hh.z * da.z + hh.w * da.w;
    agg[k] = make_float4(0.f, 0.f, 0.f, 0.f);
  }
  as[i] = s;
  ad[i] = d;
  menc[i] = enc_f32(-__builtin_inff());
  ssum[i] = 0.f;
}

// Pass 1: e = leaky_relu(as[src] + ad[dst]); store e; segment max via u32 atomicMax.
__global__ void gat_edge_logits(
    const long long* __restrict__ ei,
    const float* __restrict__ as, const float* __restrict__ ad,
    float* __restrict__ ebuf, unsigned* __restrict__ menc, int ne, int n)
{
  int idx = blockIdx.x * blockDim.x + threadIdx.x;
  if (idx >= ne + n) return;
  int s, d;
  if (idx < ne) { s = (int)ei[idx]; d = (int)ei[ne + idx]; }
  else          { s = idx - ne;     d = s; }
  float e = as[s] + ad[d];
  e = e > 0.f ? e : 0.2f * e;
  ebuf[idx] = e;
  atomicMax(&menc[d], enc_f32(e));
}

// Pass 2: ex = exp(e - m[dst]); store ex; segment sum via f32 atomicAdd.
__global__ void gat_edge_expsum(
    const long long* __restrict__ ei,
    float* __restrict__ ebuf, const unsigned* __restrict__ menc,
    float* __restrict__ ssum, int ne, int n)
{
  int idx = blockIdx.x * blockDim.x + threadIdx.x;
  if (idx >= ne + n) return;
  int d = (idx < ne) ? (int)ei[ne + idx] : (idx - ne);
  float ex = __expf(ebuf[idx] - dec_f32(menc[d]));
  ebuf[idx] = ex;
  atomicAdd(&ssum[d], ex);
}

// Pass 3: out[dst] += (ex / (s[dst]+eps)) * h[src]; 64 threads per edge.
__global__ void gat_edge_aggregate(
    const long long* __restrict__ ei,
    const float* __restrict__ ebuf, const float* __restrict__ ssum,
    const float* __restrict__ hlin, float* __restrict__ hagg, int ne, int n)
{
  long long t = (long long)blockIdx.x * blockDim.x + threadIdx.x;
  if (t >= (long long)(ne + n) * DH) return;
  int idx = (int)(t >> 6);
  int dim = (int)(t & 63);
  int s, d;
  if (idx < ne) { s = (int)ei[idx]; d = (int)ei[ne + idx]; }
  else          { s = idx - ne;     d = s; }
  float alpha = ebuf[idx] / (ssum[d] + 1e-16f);
  atomicAdd(&hagg[(size_t)d * DH + dim], alpha * hlin[(size_t)s * DH + dim]);
}

// ---------------------------------------------------------------------------
// Global mean pool + final linear head.
// ---------------------------------------------------------------------------
__global__ void gat_zero_pool(float* __restrict__ pooled, float* __restrict__ counts)
{
  int i = blockIdx.x * blockDim.x + threadIdx.x;
  if (i < NG * DH) pooled[i] = 0.f;
  if (i < NG)      counts[i] = 0.f;
}

__global__ void gat_pool(
    const long long* __restrict__ batch,
    const float* __restrict__ hagg, const float* __restrict__ b3,
    float* __restrict__ pooled, float* __restrict__ counts, int n)
{
  long long t = (long long)blockIdx.x * blockDim.x + threadIdx.x;
  if (t >= (long long)n * DH) return;
  int i = (int)(t >> 6);
  int dim = (int)(t & 63);
  int g = (int)batch[i];
  atomicAdd(&pooled[g * DH + dim], hagg[(size_t)i * DH + dim] + b3[dim]);
  if (dim == 0) atomicAdd(&counts[g], 1.0f);
}

__global__ void gat_final(
    const float* __restrict__ pooled, const float* __restrict__ counts,
    const float* __restrict__ Wl, const float* __restrict__ bl,
    float* __restrict__ out)
{
  int t = blockIdx.x * blockDim.x + threadIdx.x;
  if (t >= NG * NC) return;
  int g = t / NC, c = t % NC;
  float inv = 1.0f / fmaxf(counts[g], 1.0f);
  float acc = bl[c];
  #pragma unroll 8
  for (int k = 0; k < DH; ++k)
    acc += pooled[g * DH + k] * inv * Wl[k * NC + c];
  out[g * NC + c] = acc;
}

// ---------------------------------------------------------------------------
extern "C" void kernel_launch(void* const* d_in, const int* in_sizes, int n_in,
                              void* d_out, int out_size, void* d_ws, size_t ws_size,
                              hipStream_t stream) {
  (void)n_in; (void)out_size; (void)ws_size;

  const float*     x     = (const float*)d_in[0];       // [N,1]
  const float*     pos   = (const float*)d_in[1];       // [N,2]
  const long long* eidx  = (const long long*)d_in[2];   // [2,E] int64
  const long long* batch = (const long long*)d_in[3];   // [N]  int64
  const float* W1 = (const float*)d_in[4];
  const float* as1 = (const float*)d_in[5];
  const float* ad1 = (const float*)d_in[6];
  const float* b1  = (const float*)d_in[7];
  const float* W2 = (const float*)d_in[8];
  const float* as2 = (const float*)d_in[9];
  const float* ad2 = (const float*)d_in[10];
  const float* b2  = (const float*)d_in[11];
  const float* W3 = (const float*)d_in[12];
  const float* as3 = (const float*)d_in[13];
  const float* ad3 = (const float*)d_in[14];
  const float* b3  = (const float*)d_in[15];
  const float* Wl = (const float*)d_in[16];
  const float* bl = (const float*)d_in[17];

  const int n  = in_sizes[0];          // N = 100000 (N % 32 == 0)
  const int ne = in_sizes[2] / 2;      // E = 1600000
  const int tot = ne + n;              // edges + self loops

  // Workspace carve-up (floats).
  float* ws = (float*)d_ws;
  float*    hlin   = ws;                         // N*64
  float*    hagg   = hlin + (size_t)n * DH;      // N*64
  float*    as_buf = hagg + (size_t)n * DH;      // N
  float*    ad_buf = as_buf + n;                 // N
  unsigned* menc   = (unsigned*)(ad_buf + n);    // N
  float*    ssum   = (float*)(menc + n);         // N
  float*    ebuf   = ssum + n;                   // E+N
  float*    pooled = ebuf + tot;                 // G*64
  float*    counts = pooled + NG * DH;           // G

  const int B = 256;
  dim3 tiles1((unsigned)(n / 16), DH / 16);      // layer-1 WMMA grid (exact)
  const int gemmBlocks = n / TM;                 // layers 2/3 (exact)
  const int nodeBlocks = (n + B - 1) / B;
  const int edgeBlocks = (tot + B - 1) / B;
  const int aggBlocks  = (int)(((long long)tot * DH + B - 1) / B);
  const int poolBlocks = (int)(((long long)n * DH + B - 1) / B);

  const float* Ws[3]  = { W1, W2, W3 };
  const float* Ass[3] = { as1, as2, as3 };
  const float* Ads[3] = { ad1, ad2, ad3 };
  const float* Bs[3]  = { b1, b2, b3 };

  for (int layer = 0; layer < 3; ++layer) {
    if (layer == 0) {
      gat_linear1_wmma<<<tiles1, 32, 0, stream>>>(pos, x, Ws[0], hlin);
    } else {
      // input = previous layer's aggregation + previous bias (fused)
      gat_linear_wmma<<<gemmBlocks, B, 0, stream>>>(hagg, Bs[layer - 1],
                                                    Ws[layer], hlin);
    }
    gat_node_prep<<<nodeBlocks, B, 0, stream>>>(hlin, Ass[layer], Ads[layer],
                                                as_buf, ad_buf, menc, ssum,
                                                hagg, n);
    gat_edge_logits<<<edgeBlocks, B, 0, stream>>>(eidx, as_buf, ad_buf, ebuf,
                                                  menc, ne, n);
    gat_edge_expsum<<<edgeBlocks, B, 0, stream>>>(eidx, ebuf, menc, ssum, ne, n);
    gat_edge_aggregate<<<aggBlocks, B, 0, stream>>>(eidx, ebuf, ssum, hlin,
                                                    hagg, ne, n);
  }

  gat_zero_pool<<<(NG * DH + B - 1) / B, B, 0, stream>>>(pooled, counts);
  gat_pool<<<poolBlocks, B, 0, stream>>>(batch, hagg, b3, pooled, counts, n);
  gat_final<<<(NG * NC + B - 1) / B, B, 0, stream>>>(pooled, counts, Wl, bl,
                                                     (float*)d_out);
}